// Detection_loss_10084583211077
// MI455X (gfx1250) — compile-verified
//
#include <hip/hip_runtime.h>
#include <math.h>

#define FSIZE   76
#define NCELL   (FSIZE * FSIZE)   // 5776 cells per anchor plane
#define NA      3
#define NB      16
#define NBOX    100
#define NCLS    80
#define NCH     85
#define RECSZ   12                // floats per label record in workspace

typedef __attribute__((ext_vector_type(2))) float v2f;
typedef __attribute__((ext_vector_type(8))) float v8f;

__device__ __forceinline__ float sigm(float x) { return 1.0f / (1.0f + expf(-x)); }

// BCE with the reference's -100 log clamp: bce(0,0)==0 exactly.
__device__ __forceinline__ float bce(float x, float t) {
    float lp = fmaxf(logf(x), -100.0f);
    float ln = fmaxf(logf(1.0f - x), -100.0f);
    return -(t * lp + (1.0f - t) * ln);
}

// masked anchors: ANCHORS_PX / IMAGE_SIZE / STRIDE = px / 4864
#define MA0X (13.0f / 4864.0f)
#define MA0Y (16.0f / 4864.0f)
#define MA1X (28.0f / 4864.0f)
#define MA1Y (32.0f / 4864.0f)
#define MA2X (62.0f / 4864.0f)
#define MA2Y (35.0f / 4864.0f)

__global__ void init_out_kernel(float* __restrict__ out) {
    if (threadIdx.x < 6) out[threadIdx.x] = 0.0f;
}

// ---------------------------------------------------------------------------
// Kernel A: per-label preprocessing (CIoU argmax over 3 ref anchors, target row)
// rec layout per label: [0]=flat(int) [1]=tx [2]=tw [3]=th [4]=cls(int) [5]=scale
//                       [6..9]=label box (lx,ly,lw,lh)
// ---------------------------------------------------------------------------
__global__ void prep_labels_kernel(const float* __restrict__ labels,
                                   float* __restrict__ rec) {
    const int b = blockIdx.x;
    const int n = threadIdx.x;
    if (n >= NBOX) return;

    const float ma[3][2] = {{MA0X, MA0Y}, {MA1X, MA1Y}, {MA2X, MA2Y}};
    const float* L = labels + (size_t)(b * NBOX + n) * 5;
    const float X = L[0], Y = L[1], W = L[2], H = L[3];
    const int   cls = (int)L[4];

    const float lx = (X + W) * (1.0f / 16.0f);   // (x+w)/(STRIDE*2)
    const float ly = (Y + H) * (1.0f / 16.0f);
    const float lw = W * 0.125f;                 // w/STRIDE
    const float lh = H * 0.125f;
    const int   li = (int)lx;
    const int   lj = (int)ly;

    // argmax over 3 anchors of CIoU([0,0,lw,lh], [0,0,aw,ah]); strict > == first-max
    const float inv_pi2_4 = 4.0f / (float)(M_PI * M_PI);
    float best = -1e30f;
    int   bn   = 0;
    const float at_l = atanf(lw / fmaxf(lh, 1e-16f));
    for (int k = 0; k < 3; ++k) {
        const float aw = ma[k][0], ah = ma[k][1];
        const float iw = fminf(lw, aw), ih = fminf(lh, ah);
        const float en = (iw > 0.0f && ih > 0.0f) ? 1.0f : 0.0f;
        const float ai = iw * ih * en;
        const float iou = ai / fmaxf(lw * lh + aw * ah - ai, 1e-16f);
        const float dw = 0.5f * (lw - aw), dh = 0.5f * (lh - ah);
        const float rho2 = dw * dw + dh * dh;
        const float mw = fmaxf(lw, aw), mh = fmaxf(lh, ah);
        const float c2 = mw * mw + mh * mh;
        const float dat = at_l - atanf(aw / fmaxf(ah, 1e-16f));
        const float v = inv_pi2_4 * dat * dat;
        const float alpha = v / fmaxf(1.0f - iou + v, 1e-16f);
        const float ciou = iou - rho2 / fmaxf(c2, 1e-16f) - alpha * v;
        if (ciou > best) { best = ciou; bn = k; }
    }

    const int   flat = bn * NCELL + lj * FSIZE + li;
    const float tx   = lx - truncf(lx);
    const float aw   = ma[bn][0], ah = ma[bn][1];
    const float tw   = logf(lw / aw + 1e-16f);
    const float th   = logf(lh / ah + 1e-16f);
    const float sv   = sqrtf(2.0f - lw * lh / (float)NCELL);

    float* r = rec + (size_t)(b * NBOX + n) * RECSZ;
    r[0] = __int_as_float(flat);
    r[1] = tx; r[2] = tw; r[3] = th;
    r[4] = __int_as_float(cls);
    r[5] = sv;
    r[6] = lx; r[7] = ly; r[8] = lw; r[9] = lh;
    r[10] = 0.0f; r[11] = 0.0f;
}

// ---------------------------------------------------------------------------
// Kernel B: per-cell loss + block reduction via V_WMMA_F32_16X16X4_F32
// grid = (ceil(5776/128), 3, 16), block = 128 (4 waves of 32)
// out: [0]=loss [1]=xy [2]=wh [3]=obj [4]=cls [5]=l2
// ---------------------------------------------------------------------------
__global__ __launch_bounds__(128) void yolo_loss_kernel(
        const float* __restrict__ x,
        const float* __restrict__ rec,
        float* __restrict__ out) {
    const int tid  = threadIdx.x;
    const int cell = blockIdx.x * 128 + tid;
    const int a    = blockIdx.y;
    const int b    = blockIdx.z;
    const bool act = (cell < NCELL);

    const float ma[3][2] = {{MA0X, MA0Y}, {MA1X, MA1Y}, {MA2X, MA2Y}};

    float lxy = 0.0f, lwh = 0.0f, lobj = 0.0f, lcls = 0.0f, ll2 = 0.0f;

    const float* xb = x + (size_t)(b * NA + a) * NCH * NCELL;
    const float* rb = rec + (size_t)b * NBOX * RECSZ;   // block-uniform -> s_loads

    if (act) {
        const int i = cell % FSIZE;
        const int j = cell / FSIZE;
        // channels 0..4 (coalesced: consecutive tid -> consecutive addresses)
        const float c0 = xb[0 * NCELL + cell];
        const float c1 = xb[1 * NCELL + cell];
        const float c2 = xb[2 * NCELL + cell];
        const float c3 = xb[3 * NCELL + cell];
        const float c4 = xb[4 * NCELL + cell];
        const float s0 = sigm(c0), s1 = sigm(c1), s4 = sigm(c4);

        const float aw = ma[a][0], ah = ma[a][1];
        const float px = s0 + (float)i;
        const float py = s1 + (float)j;
        const float pw = expf(c2 * aw);   // reference multiplies INSIDE exp
        const float ph = expf(c3 * ah);
        const float pl = px - 0.5f * pw, pr = px + 0.5f * pw;
        const float pt = py - 0.5f * ph, pb = py + 0.5f * ph;
        const float pa = pw * ph;

        const int myflat = a * NCELL + cell;
        float maxiou = -1e30f;
        int   match  = -1;                 // last-writer-wins on duplicate flats
        for (int n = 0; n < NBOX; ++n) {
            const float* r = rb + n * RECSZ;
            const float LX = r[6], LY = r[7], LW = r[8], LH = r[9];
            const float tlx = fmaxf(pl, LX - 0.5f * LW);
            const float tly = fmaxf(pt, LY - 0.5f * LH);
            const float brx = fminf(pr, LX + 0.5f * LW);
            const float bry = fminf(pb, LY + 0.5f * LH);
            const float ai  = (tlx < brx && tly < bry) ? (brx - tlx) * (bry - tly) : 0.0f;
            const float iou = ai / fmaxf(pa + LW * LH - ai, 1e-16f);
            maxiou = fmaxf(maxiou, iou);
            if (__float_as_int(r[0]) == myflat) match = n;
        }

        if (match >= 0) {
            const float* r = rb + match * RECSZ;
            const float tx = r[1], tw = r[2], th = r[3], sv = r[5];
            const int   cls = __float_as_int(r[4]);
            const float w2 = sv * sv;
            // both channel 0 and 1 target tx (replicates reference)
            lxy = w2 * (bce(s0, tx) + bce(s1, tx));
            const float d2 = (c2 - tw) * sv;
            const float d3 = (c3 - th) * sv;
            lwh = 0.5f * (d2 * d2 + d3 * d3);
            lobj = -fmaxf(logf(s4), -100.0f);                 // bce(s4, 1)
            ll2  = (s0 - tx) * (s0 - tx) + (s1 - tx) * (s1 - tx)
                 + d2 * d2 + d3 * d3 + (s4 - 1.0f) * (s4 - 1.0f);
            // only matched lanes touch the 80 class channels (saves ~94% traffic)
            for (int c = 0; c < NCLS; ++c) {
                const float sc = sigm(xb[(5 + c) * NCELL + cell]);
                const float t  = (c == cls) ? 1.0f : 0.0f;
                lcls += bce(sc, t);
                const float d = sc - t;
                ll2 += d * d;
            }
        } else {
            const float objm = (maxiou > 0.5f) ? 0.0f : 1.0f;
            if (objm > 0.0f) {
                lobj = -fmaxf(logf(1.0f - s4), -100.0f);      // bce(s4, 0)
                ll2  = s4 * s4;
            }
            // objm==0: bce(0,0)==0, l2 term 0
        }
    }

    // ---- block reduction: LDS stage, then wave 0 sums 128 values per
    // component with one V_WMMA_F32_16X16X4_F32 (A = values, B = ones) ----
    __shared__ float sbuf[5][128];
    sbuf[0][tid] = lxy;
    sbuf[1][tid] = lwh;
    sbuf[2][tid] = lobj;
    sbuf[3][tid] = lcls;
    sbuf[4][tid] = ll2;
    __syncthreads();

    if (tid < 32) {                    // full wave -> EXEC all-ones for WMMA
        const int lane = tid;
        const int m    = lane & 15;
        const int kb   = (lane < 16) ? 0 : 2;   // A f32 16x4 layout (ISA 7.12.2)
        float tot03 = 0.0f;
        for (int comp = 0; comp < 5; ++comp) {
            // pre-fold 128 -> 64, place in A[m][kb..kb+1]
            const float ax = sbuf[comp][4 * m + kb]     + sbuf[comp][4 * m + kb + 64];
            const float ay = sbuf[comp][4 * m + kb + 1] + sbuf[comp][4 * m + kb + 1 + 64];
            v2f A; A.x = ax;   A.y = ay;
            v2f B; B.x = 1.0f; B.y = 1.0f;    // ones -> D[m][n] = rowsum(A, m)
            v8f C = {};
            v8f D = __builtin_amdgcn_wmma_f32_16x16x4_f32(
                        false, A, false, B, (short)0, C, false, false);
            float p = D[0] + D[1] + D[2] + D[3] + D[4] + D[5] + D[6] + D[7];
            p += __shfl_xor(p, 16, 32);       // rows 0-7 (lane n) + rows 8-15 (lane n+16)
            if (lane == 0) {
                atomicAdd(&out[1 + comp], p);
                if (comp < 4) tot03 += p;
            }
        }
        if (lane == 0) atomicAdd(&out[0], tot03);   // loss = xy + wh + obj + cls
    }
}

// ---------------------------------------------------------------------------
extern "C" void kernel_launch(void* const* d_in, const int* in_sizes, int n_in,
                              void* d_out, int out_size, void* d_ws, size_t ws_size,
                              hipStream_t stream) {
    const float* x      = (const float*)d_in[0];
    const float* labels = (const float*)d_in[1];
    float*       out    = (float*)d_out;
    float*       rec    = (float*)d_ws;    // 16*100*12 floats = 76.8 KB

    init_out_kernel<<<1, 32, 0, stream>>>(out);
    prep_labels_kernel<<<NB, 128, 0, stream>>>(labels, rec);

    dim3 grid((NCELL + 127) / 128, NA, NB);
    yolo_loss_kernel<<<grid, 128, 0, stream>>>(x, rec, out);
}